// HighOrderActivationB_16741782520154
// MI455X (gfx1250) — compile-verified
//
#include <hip/hip_runtime.h>
#include <stdint.h>

// ---------------------------------------------------------------------------
// HighOrderActivationB for MI455X (gfx1250)
//
// Memory-bound op (~184 MB moved, ~0.25 GFLOP => ~1.4 FLOP/B; HBM floor ~8us
// at 23.3 TB/s). fp32 in/out per reference. No GEMM structure worth WMMA.
// CDNA5 paths used: TDM tensor_load_to_lds (params tile -> LDS, TENSORcnt),
// s_wait_tensorcnt, ds_load_b128 LDS gathers, wide b96/b128 global traffic.
// ---------------------------------------------------------------------------

#define BATCH      8192
#define GROUPS     512
#define ARITY      3
#define OUT_DIM    8
#define NPARAM     27                      // 3^ARITY
#define ROW_FLOATS (NPARAM * OUT_DIM)      // 216 floats per group
#define G_TILE     32
#define B_TILE     32
#define BLOCK      256
#define X_STRIDE   (GROUPS * ARITY)        // 1536
#define O_STRIDE   (GROUPS * OUT_DIM)      // 4096

typedef __attribute__((ext_vector_type(4))) float        f4;
typedef __attribute__((ext_vector_type(4))) unsigned int u32x4;
typedef __attribute__((ext_vector_type(4))) int          i32x4;
typedef __attribute__((ext_vector_type(8))) int          i32x8;

__global__ __launch_bounds__(BLOCK)
void hoact_b_kernel(const float* __restrict__ X,
                    const float* __restrict__ P,
                    float* __restrict__ out)
{
    // 32 groups * 216 floats = 27,648 B of LDS (well under 320 KB/WGP)
    __shared__ __align__(16) float lds_p[G_TILE * ROW_FLOATS];

    const int g0 = blockIdx.x * G_TILE;    // first group of this tile
    const int b0 = blockIdx.y * B_TILE;    // first batch row of this tile

    // ---- Stage the (contiguous) params tile into LDS via the TDM ----------
#if __has_builtin(__builtin_amdgcn_tensor_load_to_lds)
    if ((threadIdx.x >> 5) == 0) {         // one wave issues the DMA
        const uint32_t nelem = G_TILE * ROW_FLOATS;               // 6912 f32
        const uint64_t ga = (uint64_t)(uintptr_t)(P + (size_t)g0 * ROW_FLOATS);
        const uint32_t la = (uint32_t)(uintptr_t)&lds_p[0];       // LDS byte offset

        // D# group 0: count=1 | lds_addr | global_addr[56:0] | type=2
        u32x4 d0;
        d0.x = 1u;
        d0.y = la;
        d0.z = (uint32_t)ga;
        d0.w = (uint32_t)((ga >> 32) & 0x01FFFFFFu) | (2u << 30);

        // D# group 1: data_size=4B; 1-D tile: tile_dim0 = tensor_dim0 = nelem
        i32x8 d1;
        d1[0] = (int)(2u << 16);                         // data_size=2 (4B), mask=0
        d1[1] = (int)((nelem & 0xFFFFu) << 16);          // tensor_dim0[15:0]
        d1[2] = (int)(((nelem >> 16) & 0xFFFFu)          // tensor_dim0[31:16]
                      | (1u << 16));                     // tensor_dim1 = 1
        d1[3] = (int)((nelem & 0xFFFFu) << 16);          // tile_dim0 = nelem
        d1[4] = 1;                                       // tile_dim1=1, tile_dim2=0
        d1[5] = (int)nelem;                              // tensor_dim0_stride[31:0]
        d1[6] = (int)((nelem & 0xFFFFu) << 16);          // dim0_str[47:32]=0 | dim1_str[15:0]
        d1[7] = 0;                                       // dim1_stride[47:16]=0

        i32x4 z4 = {0, 0, 0, 0};
#if __has_include(<hip/amd_detail/amd_gfx1250_TDM.h>)
        i32x8 z8 = {0, 0, 0, 0, 0, 0, 0, 0};
        __builtin_amdgcn_tensor_load_to_lds(d0, d1, z4, z4, z8, 0);
#else
        __builtin_amdgcn_tensor_load_to_lds(d0, d1, z4, z4, 0);
#endif
#if __has_builtin(__builtin_amdgcn_s_wait_tensorcnt)
        __builtin_amdgcn_s_wait_tensorcnt(0);
#else
        asm volatile("s_wait_tensorcnt 0x0" ::: "memory");
#endif
    }
#else
    {   // Fallback: cooperative vector copy (tile is contiguous in memory)
        const f4* src = (const f4*)(P + (size_t)g0 * ROW_FLOATS);
        f4*       dst = (f4*)lds_p;
        for (int i = threadIdx.x; i < (G_TILE * ROW_FLOATS) / 4; i += BLOCK)
            dst[i] = src[i];
    }
#endif
    __syncthreads();

    const int g_local = threadIdx.x & 31;         // lane -> group (coalesced)
    const int row0    = threadIdx.x >> 5;         // wave -> batch row
    const int g       = g0 + g_local;
    const f4* rowp    = (const f4*)(lds_p + g_local * ROW_FLOATS);

    #pragma unroll
    for (int p = 0; p < B_TILE / (BLOCK / 32); ++p) {
        const int b = b0 + p * (BLOCK / 32) + row0;

        // 12 B per lane, contiguous across the wave -> global_load_b96
        const float* xr = X + (size_t)b * X_STRIDE + ARITY * g;
        const float x0 = xr[0], x1 = xr[1], x2 = xr[2];

        const float a0 = __builtin_fabsf(x0);
        const float a1 = __builtin_fabsf(x1);
        const float a2 = __builtin_fabsf(x2);
        const int t0 = (x0 >= 0.f) ? 1 : -1;      // sign * 3^0
        const int t1 = (x1 >= 0.f) ? 3 : -3;      // sign * 3^1
        const int t2 = (x2 >= 0.f) ? 9 : -9;      // sign * 3^2
        const int T  = t0 + t1 + t2;

        // branchless min / max of |x| with sign-term tracking
        // (stable-argsort tie rule: min keeps earlier index, max keeps later)
        const float mn01 = fminf(a0, a1);
        const int   tn01 = (a1 < a0) ? t1 : t0;
        const float mn   = fminf(mn01, a2);
        const int   tn   = (a2 < mn01) ? t2 : tn01;

        const float mx01 = fmaxf(a0, a1);
        const int   tx01 = (a0 > a1) ? t0 : t1;
        const float mx   = fmaxf(mx01, a2);
        const int   tx   = (mx01 > a2) ? tx01 : t2;

        const float mid = (a0 + a1 + a2) - mn - mx;

        const float c0 = mn;         // coef for smallest |x|
        const float c1 = mid - mn;   // coef for middle
        const float c2 = mx - mid;   // coef for largest

        const int i_lo  = 13 + T;        // idx paired with c0
        const int i_mid = 13 + T - tn;   // idx paired with c1
        const int i_hi  = 13 + tx;       // idx paired with c2

        // 3 gathers x (2 x ds_load_b128), 6 packed FMAs
        const f4 r_lo = c0 * rowp[i_lo * 2]     + c1 * rowp[i_mid * 2]
                      + c2 * rowp[i_hi * 2];
        const f4 r_hi = c0 * rowp[i_lo * 2 + 1] + c1 * rowp[i_mid * 2 + 1]
                      + c2 * rowp[i_hi * 2 + 1];

        // 32 B per lane, contiguous across the wave -> global_store_b128 x2
        f4* op = (f4*)(out + (size_t)b * O_STRIDE + OUT_DIM * g);
        op[0] = r_lo;
        op[1] = r_hi;
    }
}

extern "C" void kernel_launch(void* const* d_in, const int* in_sizes, int n_in,
                              void* d_out, int out_size, void* d_ws, size_t ws_size,
                              hipStream_t stream)
{
    (void)in_sizes; (void)n_in; (void)out_size; (void)d_ws; (void)ws_size;
    const float* X = (const float*)d_in[0];   // [8192, 1536] f32
    const float* P = (const float*)d_in[1];   // [512, 27, 8] f32
    float*       O = (float*)d_out;           // [8192, 4096] f32

    dim3 grid(GROUPS / G_TILE, BATCH / B_TILE);   // (16, 256)
    hoact_b_kernel<<<grid, BLOCK, 0, stream>>>(X, P, O);
}